// CombinedLoss_15805479649571
// MI455X (gfx1250) — compile-verified
//
#include <hip/hip_runtime.h>
#include <hip/hip_bf16.h>
#include <math.h>

// ---------------- problem constants ----------------
#define BB 4
#define CC 8
#define HH 256
#define WW 256
#define NPIX (BB*HH*WW)            // 262144
#define SMOOTH 1e-6f

// main reduction kernel config
#define MAIN_BLOCKS 256
#define MAIN_THREADS 256           // 8 waves of 32
#define WAVES_PER_BLOCK (MAIN_THREADS/32)
#define PIX_PER_WAVE (NPIX/(MAIN_BLOCKS*WAVES_PER_BLOCK))   // 128
#define ITERS (PIX_PER_WAVE/32)    // 4
#define NPART 26                   // 8 sumP + ce + focal + 8 inter + 8 cnt

// boundary kernel config
#define SW 16                      // column-strip width
#define NSTRIP (WW/SW)             // 16
#define BND_WGS (BB*CC*NSTRIP)     // 512

typedef float v2f __attribute__((ext_vector_type(2)));
typedef float v8f __attribute__((ext_vector_type(8)));

// =====================================================================
// Kernel 1: softmax + CE/focal + per-class sums via f32 WMMA
// (soft confusion matrix:  D[1+c][n] += (t==c) * p[n],  D[0][n] += p[n])
// =====================================================================
__global__ __launch_bounds__(MAIN_THREADS)
void main_reduce_kernel(const float* __restrict__ pred,
                        const int*   __restrict__ targ,
                        float*       __restrict__ partials)
{
    __shared__ float Q[WAVES_PER_BLOCK][32][16];   // per-pixel quantity rows (B operand source)
    __shared__ int   T[WAVES_PER_BLOCK][32];       // per-pixel targets (A operand source)
    __shared__ float Cf[WAVES_PER_BLOCK][16][16];  // reconstructed D tile
    __shared__ float wavePart[WAVES_PER_BLOCK][NPART];

    const int tid  = threadIdx.x;
    const int wv   = tid >> 5;
    const int lane = tid & 31;
    const int gw   = blockIdx.x * WAVES_PER_BLOCK + wv;

    v8f acc = {};   // 16x16 f32 accumulator (8 VGPRs)

    for (int it = 0; it < ITERS; ++it) {
        const int pix = gw * PIX_PER_WAVE + it * 32 + lane;   // < NPIX by construction
        const int b   = pix >> 16;         // HH*WW = 65536
        const int hw  = pix & 0xFFFF;

        // log_softmax (max-subtracted, matches jax.nn.log_softmax)
        float x[CC];
        #pragma unroll
        for (int c = 0; c < CC; ++c) x[c] = pred[((b*CC + c) << 16) + hw];
        float mx = x[0];
        #pragma unroll
        for (int c = 1; c < CC; ++c) mx = fmaxf(mx, x[c]);
        float se = 0.f;
        #pragma unroll
        for (int c = 0; c < CC; ++c) se += expf(x[c] - mx);
        const float lse = mx + logf(se);

        const int t   = targ[pix];
        const float ce = lse - x[t];                 // -logp[t]
        const float pt = expf(x[t] - lse);           // == exp(-ce) in reference
        const float omp = 1.f - pt;
        const float focal = 0.25f * omp * omp * ce;  // alpha*(1-pt)^gamma*ce

        // stage per-pixel quantities: n=0..7 -> p[c], 8 -> ce, 9 -> focal, 10 -> 1, rest 0
        #pragma unroll
        for (int c = 0; c < CC; ++c) Q[wv][lane][c] = expf(x[c] - lse);
        Q[wv][lane][8]  = ce;
        Q[wv][lane][9]  = focal;
        Q[wv][lane][10] = 1.f;
        #pragma unroll
        for (int n = 11; n < 16; ++n) Q[wv][lane][n] = 0.f;
        T[wv][lane] = t;
        __syncthreads();

        // WMMA f32 16x16x4: A = selector [16x4], B = quantities [4x16]
        // A layout: lanes 0-15 hold K=0(v.x),K=1(v.y); lanes 16-31 hold K=2,K=3 (M = lane%16)
        // B layout mirrors the K split (N = lane%16)
        const int kb = (lane < 16) ? 0 : 2;
        const int n  = lane & 15;   // N for B
        const int m  = lane & 15;   // M for A
        #pragma unroll
        for (int g = 0; g < 8; ++g) {
            const int p0 = 4*g + kb, p1 = p0 + 1;   // wave-local pixel indices
            v2f bv; bv.x = Q[wv][p0][n]; bv.y = Q[wv][p1][n];
            const int t0 = T[wv][p0], t1 = T[wv][p1];
            v2f av;
            av.x = (m == 0) ? 1.f : ((m - 1 == t0) ? 1.f : 0.f);
            av.y = (m == 0) ? 1.f : ((m - 1 == t1) ? 1.f : 0.f);
            acc = __builtin_amdgcn_wmma_f32_16x16x4_f32(
                      false, av, false, bv, (short)0, acc, false, false);
        }
        __syncthreads();   // protect Q/T before next iteration overwrites
    }

    // reconstruct D tile: VGPR j of lane L -> M = j + 8*(L>=16), N = L%16
    #pragma unroll
    for (int j = 0; j < 8; ++j)
        Cf[wv][j + 8*(lane >> 4)][lane & 15] = acc[j];
    __syncthreads();

    if (lane < NPART) {
        float v;
        if (lane < 10)        v = Cf[wv][0][lane];                 // sumP[0..7], Sce, Sfocal
        else if (lane < 18) { const int c = lane - 10; v = Cf[wv][1 + c][c];  }  // inter[c]
        else                { const int c = lane - 18; v = Cf[wv][1 + c][10]; }  // cnt[c]
        wavePart[wv][lane] = v;
    }
    __syncthreads();

    if (tid < NPART) {                      // deterministic in-order wave fold
        float s = 0.f;
        for (int w = 0; w < WAVES_PER_BLOCK; ++w) s += wavePart[w][tid];
        partials[blockIdx.x * NPART + tid] = s;
    }
}

// =====================================================================
// Kernel 2: boundary term.  One WG per (b, c, 16-column strip).
// Exact EDT matching the reference min-plus transform with INF=1e12.
// =====================================================================
__global__ __launch_bounds__(256)
void boundary_kernel(const float* __restrict__ pred,
                     const int*   __restrict__ targ,
                     float*       __restrict__ bndPart)
{
    __shared__ unsigned int bitsPP[HH*8];          // (1-p) != 0
    __shared__ unsigned int bitsPN[HH*8];          // p != 0
    __shared__ unsigned int bitsTN[HH*8];          // onehot != 0
    __shared__ unsigned short g2s[4*HH*SW];        // row-pass sq-dist, 0xFFFF == INF
    __shared__ float red[256];
    __shared__ int anyP, anyT;

    const int tid   = threadIdx.x;
    const int wg    = blockIdx.x;
    const int strip = wg % NSTRIP;
    const int pair  = wg / NSTRIP;     // 0..31
    const int c     = pair % CC;
    const int b     = pair / CC;
    const int j0    = strip * SW;

    if (tid == 0) { anyP = 0; anyT = 0; }
    __syncthreads();

    // ---- build binary masks as bit-planes (1 = nonzero) ----
    int locP = 0, locT = 0;
    for (int widx = tid; widx < HH*8; widx += 256) {
        const int row = widx >> 3, wir = widx & 7;
        unsigned int wpp = 0, wpn = 0, wtn = 0;
        for (int bit = 0; bit < 32; ++bit) {
            const int hw = row*WW + wir*32 + bit;
            float x[CC];
            #pragma unroll
            for (int cc = 0; cc < CC; ++cc) x[cc] = pred[((b*CC + cc) << 16) + hw];
            float mx = x[0];
            #pragma unroll
            for (int cc = 1; cc < CC; ++cc) mx = fmaxf(mx, x[cc]);
            float se = 0.f;
            #pragma unroll
            for (int cc = 0; cc < CC; ++cc) se += expf(x[cc] - mx);
            const float lse = mx + logf(se);
            const float p = expf(x[c] - lse);
            if (p != 0.f)          wpn |= 1u << bit;
            if ((1.f - p) != 0.f)  wpp |= 1u << bit;
            if (targ[(b << 16) + hw] == c) wtn |= 1u << bit;
        }
        bitsPP[widx] = wpp; bitsPN[widx] = wpn; bitsTN[widx] = wtn;
        locP |= (wpn != 0u); locT |= (wtn != 0u);
    }
    if (locP) atomicOr(&anyP, 1);
    if (locT) atomicOr(&anyT, 1);
    __syncthreads();

    // ---- row pass: nearest zero-pixel in row -> squared dist (u16) ----
    // plane 0: pred pos  (zeros where bitsPP==0)
    // plane 1: pred neg  (zeros where bitsPN==0)
    // plane 2: targ pos  (zeros where bitsTN==1)  <- nz=(1-onehot), zero where onehot==1
    // plane 3: targ neg  (zeros where bitsTN==0)
    for (int task = tid; task < 4*HH; task += 256) {
        const int plane = task >> 8;
        const int row   = task & 255;
        const unsigned int* src = (plane == 0) ? bitsPP : (plane == 1) ? bitsPN : bitsTN;
        const bool inv = (plane != 2);
        unsigned int zw[8];
        #pragma unroll
        for (int i = 0; i < 8; ++i) {
            unsigned int w = src[row*8 + i];
            zw[i] = inv ? ~w : w;           // 1 == zero-pixel
        }
        for (int jj = 0; jj < SW; ++jj) {
            const int j = j0 + jj;
            int dmin = 1 << 29;
            {   // nearest set bit at index <= j
                int wi = j >> 5;
                unsigned int m = zw[wi] & (0xFFFFFFFFu >> (31 - (j & 31)));
                for (;;) {
                    if (m) { dmin = j - (wi*32 + 31 - __builtin_clz(m)); break; }
                    if (wi == 0) break;
                    --wi; m = zw[wi];
                }
            }
            {   // nearest set bit at index >= j
                int wi = j >> 5;
                unsigned int m = zw[wi] & (0xFFFFFFFFu << (j & 31));
                for (;;) {
                    if (m) { int d = (wi*32 + __builtin_ctz(m)) - j; if (d < dmin) dmin = d; break; }
                    if (wi == 7) break;
                    ++wi; m = zw[wi];
                }
            }
            g2s[plane*HH*SW + row*SW + jj] =
                (dmin <= 255) ? (unsigned short)(dmin*dmin) : (unsigned short)0xFFFF;
        }
    }
    __syncthreads();

    // ---- column pass + sdf + |pred_sdf - targ_sdf| accumulation ----
    const bool eP = (anyP == 0);
    const bool eT = (anyT == 0);
    float accum = 0.f;
    for (int q = tid; q < HH*SW; q += 256) {
        const int i  = q / SW;
        const int jj = q % SW;
        int dm0 = 0x7FFFFFFF, dm1 = 0x7FFFFFFF, dm2 = 0x7FFFFFFF, dm3 = 0x7FFFFFFF;
        for (int k = 0; k < HH; ++k) {
            const int dh = k - i;
            const int d2 = dh*dh;
            const int base = k*SW + jj;
            int v;
            v = g2s[0*HH*SW + base]; if (v != 0xFFFF) { const int cd = v + d2; if (cd < dm0) dm0 = cd; }
            v = g2s[1*HH*SW + base]; if (v != 0xFFFF) { const int cd = v + d2; if (cd < dm1) dm1 = cd; }
            v = g2s[2*HH*SW + base]; if (v != 0xFFFF) { const int cd = v + d2; if (cd < dm2) dm2 = cd; }
            v = g2s[3*HH*SW + base]; if (v != 0xFFFF) { const int cd = v + d2; if (cd < dm3) dm3 = cd; }
        }
        const float INFV = 1e12f;
        const float e0 = sqrtf((dm0 == 0x7FFFFFFF) ? INFV : (float)dm0);
        const float e1 = sqrtf((dm1 == 0x7FFFFFFF) ? INFV : (float)dm1);
        const float e2 = sqrtf((dm2 == 0x7FFFFFFF) ? INFV : (float)dm2);
        const float e3 = sqrtf((dm3 == 0x7FFFFFFF) ? INFV : (float)dm3);
        const float sp = eP ? 3.0f : (fminf(fmaxf(e0 - e1, -5.f), 5.f) / 5.0f);
        const float st = eT ? 3.0f : (fminf(fmaxf(e2 - e3, -5.f), 5.f) / 5.0f);
        accum += fabsf(sp - st);
    }
    red[tid] = accum;
    __syncthreads();
    for (int s = 128; s > 0; s >>= 1) {
        if (tid < s) red[tid] += red[tid + s];
        __syncthreads();
    }
    if (tid == 0) bndPart[wg] = red[0];
}

// =====================================================================
// Kernel 3: deterministic final fold + loss formula
// =====================================================================
__global__ __launch_bounds__(32)
void final_kernel(const float* __restrict__ partials,
                  const float* __restrict__ bndPart,
                  float* __restrict__ out)
{
    __shared__ float s[NPART];
    __shared__ float bsum;
    const int tid = threadIdx.x;
    if (tid < NPART) {
        float a = 0.f;
        for (int bidx = 0; bidx < MAIN_BLOCKS; ++bidx) a += partials[bidx*NPART + tid];
        s[tid] = a;
    }
    if (tid == 31) {
        float a = 0.f;
        for (int i = 0; i < BND_WGS; ++i) a += bndPart[i];
        bsum = a;
    }
    __syncthreads();
    if (tid == 0) {
        const float Np = (float)NPIX;
        const float ce    = s[8] / Np;
        const float focal = s[9] / Np;
        float dice = 0.f, tver = 0.f, lov = 0.f;
        for (int c = 0; c < CC; ++c) {
            const float sp = s[c];            // sum of p over all pixels
            const float in = s[10 + c];       // sum of p where target==c
            const float cn = s[18 + c];       // count of target==c
            dice += 1.f - (2.f*in + SMOOTH) / (sp + cn + SMOOTH);
            const float fp = sp - in, fn = cn - in;
            tver += 1.f - (in + SMOOTH) / (in + 0.3f*fp + 0.7f*fn + SMOOTH);
            if (cn > 0.f) lov += cn * (cn + sp - 2.f*in);   // degenerate Lovasz
        }
        dice /= (float)CC;
        tver /= (float)CC;
        lov  /= (float)BB;
        const float bnd = bsum / (float)(CC * NPIX);
        out[0] = ce + 0.3f*dice + 0.3f*focal + 0.2f*tver + 0.1f*bnd + 0.1f*lov;
    }
}

// =====================================================================
extern "C" void kernel_launch(void* const* d_in, const int* in_sizes, int n_in,
                              void* d_out, int out_size, void* d_ws, size_t ws_size,
                              hipStream_t stream)
{
    (void)in_sizes; (void)n_in; (void)out_size; (void)ws_size;
    const float* pred = (const float*)d_in[0];
    const int*   targ = (const int*)d_in[1];
    float* ws       = (float*)d_ws;
    float* mainPart = ws;                         // MAIN_BLOCKS * NPART floats
    float* bndPart  = ws + MAIN_BLOCKS * NPART;   // BND_WGS floats

    main_reduce_kernel<<<MAIN_BLOCKS, MAIN_THREADS, 0, stream>>>(pred, targ, mainPart);
    boundary_kernel<<<BND_WGS, 256, 0, stream>>>(pred, targ, bndPart);
    final_kernel<<<1, 32, 0, stream>>>(mainPart, bndPart, (float*)d_out);
}